// Model_51453708206354
// MI455X (gfx1250) — compile-verified
//
#include <hip/hip_runtime.h>
#include <cstdint>
#include <cstddef>

// Problem constants (fixed by the reference).
#define B_   8
#define S_   2048
#define H_   2048
#define TPB  128         // threads per block (4 wave32)
#define TS   8           // time steps staged per LDS tile
#define NBUF 4           // LDS pipeline depth: 3 tiles (48 async ops) in flight <= 63 cap
#define PF   8           // L2 prefetch distance in tiles (64 steps ~ 2500 cycles ahead)

#define LOG2E 1.4426950408889634f
#define LN2   0.6931471805599453f

// ---- CDNA5 async global->LDS copy (ASYNCcnt path) ----------------------
// Builtin signature (from hipcc diagnostics): param 1 is a non-const
// address_space(1) int* (global source); param 2 is the LDS dest.
#define AS_GLOBAL(p) ((__attribute__((address_space(1))) int*)((void*)(p)))
#define AS_LDS(p)    ((__attribute__((address_space(3))) int*)((void*)(p)))

#if defined(__AMDGCN__) && __has_builtin(__builtin_amdgcn_global_load_async_to_lds_b32)
#define ASYNC_CP_B32(gsrc, ldst) \
  __builtin_amdgcn_global_load_async_to_lds_b32(AS_GLOBAL(gsrc), AS_LDS(ldst), 0, 0)
#else
#define ASYNC_CP_B32(gsrc, ldst)                                                         \
  asm volatile("global_load_async_to_lds_b32 %0, %1, off"                                \
               :: "v"((uint32_t)(uintptr_t)(__attribute__((address_space(3))) float*)(ldst)), \
                  "v"((uint64_t)(uintptr_t)(gsrc))                                       \
               : "memory")
#endif

#if defined(__AMDGCN__) && __has_builtin(__builtin_amdgcn_s_wait_asynccnt)
#define WAIT_ASYNC(n)                                    \
  do {                                                   \
    __builtin_amdgcn_s_wait_asynccnt((short)(n));        \
    asm volatile("" ::: "memory");                       \
  } while (0)
#else
#define WAIT_ASYNC(n) asm volatile("s_wait_asynccnt %0" :: "i"(n) : "memory")
#endif

// Stage one TS-step tile of k and v rows into LDS buffer `bufi` (16 async ops).
// Each thread fetches only its own channel's column -> no cross-thread LDS
// sharing -> no barriers, only s_wait_asynccnt.
#define ISSUE_TILE(bufi, ti)                                                              \
  do {                                                                                    \
    _Pragma("unroll")                                                                     \
    for (int ts_ = 0; ts_ < TS; ++ts_) {                                                  \
      ASYNC_CP_B32(kbase + (size_t)((ti) * TS + ts_) * H_, &tile[bufi][0][ts_][tid]);     \
      ASYNC_CP_B32(vbase + (size_t)((ti) * TS + ts_) * H_, &tile[bufi][1][ts_][tid]);     \
    }                                                                                     \
  } while (0)

// HBM -> L2 staging, PF tiles ahead. global_prefetch_b8 uses no counters,
// so this depth is free; 192MB L2 holds the window trivially.
#define PREFETCH_TILE(ti)                                                                 \
  do {                                                                                    \
    if ((ti) < S_ / TS) {                                                                 \
      _Pragma("unroll")                                                                   \
      for (int ts_ = 0; ts_ < TS; ++ts_) {                                                \
        __builtin_prefetch(kbase + (size_t)((ti) * TS + ts_) * H_, 0, 1);                 \
        __builtin_prefetch(vbase + (size_t)((ti) * TS + ts_) * H_, 0, 1);                 \
      }                                                                                   \
    }                                                                                     \
  } while (0)

#define COMPUTE_TILE(bufi)                                                \
  do {                                                                    \
    _Pragma("unroll")                                                     \
    for (int ts_ = 0; ts_ < TS; ++ts_) {                                  \
      const float kt2 = tile[bufi][0][ts_][tid] * LOG2E;                  \
      const float vt  = tile[bufi][1][ts_][tid];                          \
      /* output branch: time_first bonus, no decay */                     \
      const float a    = kt2 + tf2;                                       \
      const float mo   = fmaxf(m2, a);                                    \
      const float e1o  = __builtin_amdgcn_exp2f(m2 - mo);                 \
      const float e2o  = __builtin_amdgcn_exp2f(a - mo);                  \
      const float onum = fmaf(e1o, num, e2o * vt);                        \
      const float oden = fmaf(e1o, den, e2o);                             \
      __builtin_nontemporal_store(onum * __builtin_amdgcn_rcpf(oden), obase); \
      obase += H_;                                                        \
      /* state branch: decay, no bonus */                                 \
      const float mt  = m2 + td2;                                         \
      const float ms  = fmaxf(mt, kt2);                                   \
      const float e1s = __builtin_amdgcn_exp2f(mt - ms);                  \
      const float e2s = __builtin_amdgcn_exp2f(kt2 - ms);                 \
      num = fmaf(e1s, num, e2s * vt);                                     \
      den = fmaf(e1s, den, e2s);                                          \
      m2  = ms;                                                           \
    }                                                                     \
  } while (0)

__global__ __launch_bounds__(TPB) void wkv_scan_kernel(
    const float* __restrict__ time_decay,
    const float* __restrict__ k,
    const float* __restrict__ time_first,
    const float* __restrict__ v,
    const float* __restrict__ max_state,
    const float* __restrict__ num_state,
    const float* __restrict__ den_state,
    float* __restrict__ out)
{
  // [buffer][k/v][timestep][lane] = 4*2*8*128*4 = 32 KB of the 320 KB WGP LDS
  __shared__ float tile[NBUF][2][TS][TPB];

  const int tid = threadIdx.x;
  const int c   = blockIdx.x * TPB + tid;   // channel id in [0, B*H)
  const int b   = c / H_;
  const int h   = c % H_;

  // Work in the log2 domain: exp(x) == exp2(x*log2e); e1/e2 ratios are
  // unchanged, so num/den/out are the same math with raw v_exp_f32 and
  // no per-exp multiply.
  const float td2 = -__expf(time_decay[h]) * LOG2E;   // -exp(w) scaled
  const float tf2 = time_first[h] * LOG2E;
  float m2  = max_state[c] * LOG2E;
  float num = num_state[c];
  float den = den_state[c];

  const float* kbase = k   + ((size_t)b * S_) * H_ + h;
  const float* vbase = v   + ((size_t)b * S_) * H_ + h;
  float*       obase = out + ((size_t)b * S_) * H_ + h;

  const int NTILES = S_ / TS;               // 256

  // Prologue: fill the async pipeline with tiles 0..NBUF-2 (48 ops in flight).
  ISSUE_TILE(0, 0);
  ISSUE_TILE(1, 1);
  ISSUE_TILE(2, 2);

  // Steady state: wait(32) guarantees tile i complete (only i+1, i+2 remain),
  // then top the pipeline back up with tile i+3 and prefetch tile i+PF to L2.
  for (int i = 0; i < NTILES - 2; ++i) {
    WAIT_ASYNC(2 * TS * (NBUF - 2));        // <= 32 outstanding
    if (i + NBUF - 1 < NTILES) {
      ISSUE_TILE((i + NBUF - 1) & (NBUF - 1), i + NBUF - 1);
    }
    PREFETCH_TILE(i + PF);
    COMPUTE_TILE(i & (NBUF - 1));
  }
  // Tail: step the wait level down so each remaining tile is truly complete.
  WAIT_ASYNC(2 * TS);                       // <= 16: tile NTILES-2 done
  COMPUTE_TILE((NTILES - 2) & (NBUF - 1));
  WAIT_ASYNC(0);                            // drain: tile NTILES-1 done
  COMPUTE_TILE((NTILES - 1) & (NBUF - 1));

  // Final states, converted back out of the log2 domain for m.
  float* st = out + (size_t)B_ * S_ * H_;
  st[c]               = m2 * LN2;
  st[B_ * H_ + c]     = num;
  st[2 * B_ * H_ + c] = den;
}

extern "C" void kernel_launch(void* const* d_in, const int* in_sizes, int n_in,
                              void* d_out, int out_size, void* d_ws, size_t ws_size,
                              hipStream_t stream) {
  (void)in_sizes; (void)n_in; (void)out_size; (void)d_ws; (void)ws_size;
  const float* time_decay = (const float*)d_in[0];
  const float* key        = (const float*)d_in[1];
  const float* time_first = (const float*)d_in[2];
  const float* value      = (const float*)d_in[3];
  const float* max_state  = (const float*)d_in[4];
  const float* num_state  = (const float*)d_in[5];
  const float* den_state  = (const float*)d_in[6];
  float* out = (float*)d_out;

  dim3 grid((B_ * H_) / TPB);
  dim3 block(TPB);
  hipLaunchKernelGGL(wkv_scan_kernel, grid, block, 0, stream,
                     time_decay, key, time_first, value,
                     max_state, num_state, den_state, out);
}